// NeuralSplineBase_68659347193880
// MI455X (gfx1250) — compile-verified
//
#include <hip/hip_runtime.h>

typedef __attribute__((ext_vector_type(2))) float v2f;
typedef __attribute__((ext_vector_type(8))) float v8f;

#define N_KNOTS 10
#define B_DIM 8
#define E_DIM 3
#define C_DIM 3
#define H_DIM 768
#define W_DIM 768
#define HW (H_DIM * W_DIM)
#define NBEC 72            /* 8*3*3 */
#define NSEG 9
#define NVALS_LUT 255
#define STEPF (1.0f / 9.0f)

// ---------------------------------------------------------------------------
// Kernel 1: spline coefficient solve.
//   ys_total = ys + identity ramp;  M = MAT @ ys_total^T  via chained
//   V_WMMA_F32_16X16X4_F32 (K=16 in 4 steps, 8 waves x one 16x16 tile).
//   Then a,b,c,d per segment -> coef[(b*3+c)*27 + e*9 + i] as float4.
// ---------------------------------------------------------------------------
__global__ __launch_bounds__(256) void spline_coeff_kernel(
    const float* __restrict__ ys, float4* __restrict__ coef) {
  __shared__ float MATp[16][16];    // padded 10x10 spline matrix
  __shared__ float ysT[16][128];    // ys_total transposed, K=16 x N=128 (pad)
  __shared__ float Mbuf[16][128];   // M = MAT @ ysT
  __shared__ float w8[8][16];       // Gauss-Jordan workspace (8x8 | I)

  const int tid = threadIdx.x;

  // zero-fill padded operands
  for (int i = tid; i < 16 * 16; i += 256) ((float*)MATp)[i] = 0.0f;
  for (int i = tid; i < 16 * 128; i += 256) ((float*)ysT)[i] = 0.0f;

  // ys_total[bec][j] = ys + j/9, stored transposed ysT[j][bec]
  for (int i = tid; i < NBEC * N_KNOTS; i += 256) {
    int bec = i / N_KNOTS, j = i % N_KNOTS;
    ysT[j][bec] = ys[bec * N_KNOTS + j] + (float)j * STEPF;
  }
  __syncthreads();

  if (tid == 0) {
    // Invert the [1 4 1] tridiagonal 8x8 (diag-dominant -> no pivoting)
    for (int r = 0; r < 8; ++r)
      for (int q = 0; q < 16; ++q)
        w8[r][q] = (q < 8)
                       ? ((q == r) ? 4.0f : ((q == r - 1 || q == r + 1) ? 1.0f : 0.0f))
                       : ((q - 8 == r) ? 1.0f : 0.0f);
    for (int p = 0; p < 8; ++p) {
      float piv = 1.0f / w8[p][p];
      for (int q = 0; q < 16; ++q) w8[p][q] *= piv;
      for (int r = 0; r < 8; ++r)
        if (r != p) {
          float f = w8[r][p];
          for (int q = 0; q < 16; ++q) w8[r][q] -= f * w8[p][q];
        }
    }
    // MAT = vstack(0, (6/step^2)*inv, 0) @ Band[1,-2,1]; 6/step^2 = 486
    for (int i = 1; i <= 8; ++i)
      for (int j = 0; j < N_KNOTS; ++j) {
        float s = 0.0f;
        if (j < 8) s += w8[i - 1][8 + j];
        if (j - 1 >= 0 && j - 1 < 8) s -= 2.0f * w8[i - 1][8 + (j - 1)];
        if (j - 2 >= 0 && j - 2 < 8) s += w8[i - 1][8 + (j - 2)];
        MATp[i][j] = 486.0f * s;
      }
  }
  __syncthreads();

  // --- WMMA phase: every wave owns one 16x16 output tile (cols 72..127 pad)
  {
    const int wave = tid >> 5;
    const int lane = tid & 31;
    const int m = lane & 15;       // A row / D col-within-tile
    const int half = lane >> 4;
    const int n = wave * 16 + m;   // global column (bec, padded)
    v8f acc = {};
#pragma unroll
    for (int kc = 0; kc < 4; ++kc) {
      const int kb = kc * 4;
      // 32-bit A 16x4 layout: K = vgpr + 2*half  (lanes 0-15: K0/K1, 16-31: K2/K3)
      v2f av, bv;
      av.x = MATp[m][kb + 2 * half + 0];
      av.y = MATp[m][kb + 2 * half + 1];
      bv.x = ysT[kb + 2 * half + 0][n];
      bv.y = ysT[kb + 2 * half + 1][n];
      acc = __builtin_amdgcn_wmma_f32_16x16x4_f32(
          /*neg_a=*/false, av, /*neg_b=*/false, bv,
          /*c_mod=*/(short)0, acc, /*reuse_a=*/false, /*reuse_b=*/false);
    }
    // 32-bit C/D 16x16 layout: row = vgpr + 8*half
#pragma unroll
    for (int v = 0; v < 8; ++v) Mbuf[v + 8 * half][n] = acc[v];
  }
  __syncthreads();

  // --- per-segment coefficients: a=(M1-M0)*1.5, b=M0/2,
  //     c=9*(y1-y0)-(M1+2*M0)/54, d=y0          (h = 1/9)
  for (int idx = tid; idx < NBEC * NSEG; idx += 256) {
    int bec = idx / NSEG, i = idx % NSEG;
    float M0 = Mbuf[i][bec], M1 = Mbuf[i + 1][bec];
    float y0 = ysT[i][bec], y1 = ysT[i + 1][bec];
    float ca = (M1 - M0) * 1.5f;
    float cb = M0 * 0.5f;
    float cc = (y1 - y0) * 9.0f - (M1 + 2.0f * M0) * (1.0f / 54.0f);
    float cd = y0;
    int b = bec / 9, e = (bec % 9) / 3, c = bec % 3;
    coef[(b * 3 + c) * 27 + e * 9 + i] = make_float4(ca, cb, cc, cd);
  }
}

// ---------------------------------------------------------------------------
// Kernel 2: streaming per-pixel eval. 1 float4 in -> 3 float4 out (E experts).
// Pure bandwidth kernel: ~227MB total traffic, ~10us at 23.3 TB/s.
// ---------------------------------------------------------------------------
__global__ __launch_bounds__(256) void spline_eval_kernel(
    const float4* __restrict__ batch4, const float4* __restrict__ coef,
    float4* __restrict__ out4) {
  __shared__ float4 lc[27];                 // 3 experts x 9 segments
  const int bc = blockIdx.y;                // b*3 + c
  const int b = bc / 3;
  const int c = bc % 3;
  const int tid = threadIdx.x;
  const int planeQ = HW / 4;
  const int pix = blockIdx.x * 256 + tid;

  const float4* inp = batch4 + (size_t)bc * planeQ + pix;
  __builtin_prefetch(inp, 0, 0);            // global_prefetch_b8: overlap with LDS stage

  if (tid < 27) lc[tid] = coef[bc * 27 + tid];
  __syncthreads();

  float4 x = *inp;
  float xs[4] = {x.x, x.y, x.z, x.w};
  int xi[4];
  float xf[4];
#pragma unroll
  for (int k = 0; k < 4; ++k) {
    float t = fminf(fmaxf(xs[k] * 9.0f, 0.0f), 8.0f);
    xi[k] = (int)t;                               // floor for t >= 0
    xf[k] = fmaf((float)-xi[k], STEPF, xs[k]);    // x - xi*step
  }
#pragma unroll
  for (int e = 0; e < 3; ++e) {
    float rr[4];
#pragma unroll
    for (int k = 0; k < 4; ++k) {
      float4 kf = lc[e * 9 + xi[k]];              // ds_load_b128 gather
      rr[k] = fmaf(fmaf(fmaf(kf.x, xf[k], kf.y), xf[k], kf.z), xf[k], kf.w);
    }
    out4[((size_t)(b * 3 + e) * 3 + c) * planeQ + pix] =
        make_float4(rr[0], rr[1], rr[2], rr[3]);
  }
}

// ---------------------------------------------------------------------------
// Kernel 3: 255-entry LUT per (b,e,c) — tiny.
// ---------------------------------------------------------------------------
__global__ __launch_bounds__(256) void spline_lut_kernel(
    const float4* __restrict__ coef, float* __restrict__ outs) {
  const int bec = blockIdx.x;               // b*9 + e*3 + c
  const int p = threadIdx.x;
  if (p >= NVALS_LUT) return;
  const int b = bec / 9, e = (bec % 9) / 3, c = bec % 3;
  const float x = (float)p * (1.0f / 255.0f);
  float t = fminf(x * 9.0f, 8.0f);
  int xi = (int)t;
  float xf = fmaf((float)-xi, STEPF, x);
  float4 kf = coef[(b * 3 + c) * 27 + e * 9 + xi];
  outs[bec * NVALS_LUT + p] =
      fmaf(fmaf(fmaf(kf.x, xf, kf.y), xf, kf.z), xf, kf.w);
}

extern "C" void kernel_launch(void* const* d_in, const int* in_sizes, int n_in,
                              void* d_out, int out_size, void* d_ws, size_t ws_size,
                              hipStream_t stream) {
  const float4* batch4 = (const float4*)d_in[0];   // (8,3,768,768) f32
  const float* ys = (const float*)d_in[1];         // (8,3,3,10)    f32
  float4* coef = (float4*)d_ws;                    // 24*27 float4 = 10368 B
  float* out = (float*)d_out;                      // (8,3,3,768,768) ++ (8,3,3,255)

  spline_coeff_kernel<<<1, 256, 0, stream>>>(ys, coef);

  dim3 grid(HW / 4 / 256, B_DIM * C_DIM);          // (576, 24)
  spline_eval_kernel<<<grid, 256, 0, stream>>>(batch4, coef, (float4*)out);

  spline_lut_kernel<<<NBEC, 256, 0, stream>>>(
      coef, out + (size_t)B_DIM * E_DIM * C_DIM * HW);
}